// TransitionGNN_39195871543399
// MI455X (gfx1250) — compile-verified
//
#include <hip/hip_runtime.h>

// ---------------------------------------------------------------------------
// Problem constants (from reference): T=16, N=128, D=64, H=256, OUT=1
//   nodes  = states.reshape(2048, 64)
//   P = nodes @ W1[0:64 ,:] + b1   -> [2048, 256]
//   Q = nodes @ W1[64:128,:]       -> [2048, 256]
//   out[t,i,jidx] = relu(P[t*128+i] + Q[t*128+j]) . W2 + b2,  j = jidx + (jidx>=i)
// ---------------------------------------------------------------------------

typedef __attribute__((ext_vector_type(2))) float v2f;
typedef __attribute__((ext_vector_type(8))) float v8f;

#define TT   16
#define NN   128
#define DD   64
#define HH   256
#define MTOT 2048          // T*N nodes
#define NPQ  512           // P(256) | Q(256) columns
#define EPT  (NN * (NN - 1))   // 16256 edges per timestep
#define ETOT (TT * EPT)        // 260096 edges

// ---------------------------------------------------------------------------
// Stage 1: PQ[2048,512] = nodes[2048,64] @ B[64,512]  (+ b1 on the P half)
//   B[k, n] = (n < 256) ? W1[k, n] : W1[k + 64, n - 256]
// One wave computes a 16(M) x 64(N) strip with 4 f32 WMMA accumulators.
// Block = 256 threads = 8 waves = the 8 N-groups of one M-tile.
// ---------------------------------------------------------------------------
__global__ __launch_bounds__(256) void pq_gemm_kernel(
    const float* __restrict__ nodes,   // [2048, 64]
    const float* __restrict__ W1,      // [128, 256] row-major
    const float* __restrict__ b1,      // [256]
    float* __restrict__ PQ)            // [2048, 512]
{
    const int lane  = threadIdx.x & 31;
    const int wave  = threadIdx.x >> 5;      // 0..7 -> N group of 64
    const int half  = lane >> 4;             // ISA: lanes 16-31 hold upper K pair
    const int lm    = lane & 15;             // M (A) / N (B,C,D) within tile
    const int Mbase = blockIdx.x * 16;
    const int Nbase = wave * 64;

    const float* Arow = nodes + (Mbase + lm) * DD;

    v8f acc[4] = {v8f{}, v8f{}, v8f{}, v8f{}};

#pragma unroll
    for (int kk = 0; kk < DD / 4; ++kk) {
        const int k0 = kk * 4 + half * 2;
        // A fragment: 16x4 f32, lane lm holds row M=lm, K = {k0, k0+1}
        const v2f a = *(const v2f*)(Arow + k0);     // 8B aligned (k0 even)

#pragma unroll
        for (int nt = 0; nt < 4; ++nt) {
            const int n = Nbase + nt * 16 + lm;
            // Column half is uniform across the wave's 16-col subtile
            // (Nbase+nt*16 is 16-aligned and 256 is a multiple of 16).
            const float* Bp = (n < HH) ? (W1 + k0 * HH + n)
                                       : (W1 + (k0 + DD) * HH + (n - HH));
            v2f b;
            b.x = Bp[0];      // B[k0  , n]
            b.y = Bp[HH];     // B[k0+1, n]  (next W1 row, stride 256 floats)

            acc[nt] = __builtin_amdgcn_wmma_f32_16x16x4_f32(
                /*neg_a=*/false, a, /*neg_b=*/false, b,
                /*c_mod=*/(short)0, acc[nt],
                /*reuse_a=*/false, /*reuse_b=*/false);
        }
    }

    // Epilogue: D layout — VGPR v holds M = v + half*8, N = lm.
    // Fold b1 into the P half (n < 256) so stage 2 skips the bias load.
#pragma unroll
    for (int nt = 0; nt < 4; ++nt) {
        const int   n    = Nbase + nt * 16 + lm;
        const float bias = (n < HH) ? b1[n] : 0.0f;
#pragma unroll
        for (int v = 0; v < 8; ++v) {
            const int m = Mbase + v + half * 8;
            PQ[m * NPQ + n] = acc[nt][v] + bias;
        }
    }
}

// ---------------------------------------------------------------------------
// Stage 2: one wave per edge.
//   s = sum_h relu(Pb[row][h] + Q[col][h]) * W2[h];  out[e] = s + b2
// Lanes cover h with contiguous float4 chunks (lane-stride 16B, coalesced),
// then a wave32 xor-butterfly reduction.
// ---------------------------------------------------------------------------
__global__ __launch_bounds__(256) void edge_mlp_kernel(
    const float* __restrict__ PQ,    // [2048, 512] (P half already has +b1)
    const float* __restrict__ W2,    // [256]
    const float* __restrict__ b2,    // [1]
    float* __restrict__ out)         // [260096] == [T, N, N-1]
{
    const int lane = threadIdx.x & 31;
    const int e    = blockIdx.x * 8 + (threadIdx.x >> 5);

    const int t    = e / EPT;
    int       rem  = e - t * EPT;
    const int i    = rem / (NN - 1);
    const int jidx = rem - i * (NN - 1);
    const int j    = jidx + (jidx >= i ? 1 : 0);

    const float4* P4 = (const float4*)(PQ + (t * NN + i) * NPQ);        // 2KB aligned
    const float4* Q4 = (const float4*)(PQ + (t * NN + j) * NPQ + HH);   // 1KB aligned
    const float4* W4 = (const float4*)W2;

    float s = 0.0f;
#pragma unroll
    for (int u = 0; u < 2; ++u) {
        const int idx = u * 32 + lane;          // float4 index, 0..63 -> h = idx*4
        const float4 p = P4[idx];
        const float4 q = Q4[idx];
        const float4 w = W4[idx];
        float h0 = fmaxf(p.x + q.x, 0.0f);
        float h1 = fmaxf(p.y + q.y, 0.0f);
        float h2 = fmaxf(p.z + q.z, 0.0f);
        float h3 = fmaxf(p.w + q.w, 0.0f);
        s = fmaf(h0, w.x, s);
        s = fmaf(h1, w.y, s);
        s = fmaf(h2, w.z, s);
        s = fmaf(h3, w.w, s);
    }

#pragma unroll
    for (int off = 16; off > 0; off >>= 1)
        s += __shfl_xor(s, off, 32);

    if (lane == 0)
        out[e] = s + b2[0];
}

// ---------------------------------------------------------------------------
// Launch: inputs are {states, action(unused), W1, b1, W2, b2}, all fp32.
// Workspace: PQ = 2048*512*4 = 4 MB at the front of d_ws.
// ---------------------------------------------------------------------------
extern "C" void kernel_launch(void* const* d_in, const int* in_sizes, int n_in,
                              void* d_out, int out_size, void* d_ws, size_t ws_size,
                              hipStream_t stream) {
    const float* states = (const float*)d_in[0];
    const float* W1     = (const float*)d_in[2];
    const float* b1     = (const float*)d_in[3];
    const float* W2     = (const float*)d_in[4];
    const float* b2     = (const float*)d_in[5];
    float*       PQ     = (float*)d_ws;       // [2048, 512] fp32 = 4 MB
    float*       outp   = (float*)d_out;      // [260096] fp32

    // Stage 1: node-level WMMA GEMM (128 M-tiles x 8 waves/N-groups each)
    pq_gemm_kernel<<<MTOT / 16, 256, 0, stream>>>(states, W1, b1, PQ);

    // Stage 2: 260096 edges, 8 edge-waves per 256-thread block
    edge_mlp_kernel<<<ETOT / 8, 256, 0, stream>>>(PQ, W2, b2, outp);
}